// CapsuleLayer_75514114998376
// MI455X (gfx1250) — compile-verified
//
#include <hip/hip_runtime.h>
#include <hip/hip_bf16.h>
#include <math.h>

typedef __attribute__((ext_vector_type(2))) float v2f;
typedef __attribute__((ext_vector_type(4))) float v4f;
typedef __attribute__((ext_vector_type(8))) float v8f;

#define B_    32
#define SEQ   2048
#define IND   128
#define ONUM  16
#define ODIM  32
#define CHUNKS 16
#define ROWS_PER_CHUNK (SEQ / CHUNKS)          // 128
#define WAVES 4
#define TILES_PER_WAVE (ROWS_PER_CHUNK / (16 * WAVES)) // 2
#define XT_STRIDE 132                          // 16x128 tile padded: 132 % 64 = 4 -> conflict-free
#define CS_STRIDE 17

// workspace layout (in floats)
#define OFF_XSUM 0                                   // [CHUNKS][B][128]
#define OFF_V    (CHUNKS * B_ * IND)                 // [B][16][128]
#define OFF_YP   (OFF_V + B_ * ONUM * IND)           // [CHUNKS][B][16][128]

// ---------------------------------------------------------------------------
// Kernel 1: per-chunk partial sum of x over the sequence dim (deterministic,
// no atomics). xsum_part[chunk][b][i] = sum_{s in chunk} x[b][s][i]
// ---------------------------------------------------------------------------
__global__ __launch_bounds__(128) void xsum_kernel(const float* __restrict__ x,
                                                   float* __restrict__ xsp) {
  const int b = blockIdx.x;
  const int c = blockIdx.y;
  const int i = threadIdx.x;  // 128 features, coalesced
  const float* xb = x + ((size_t)b * SEQ + (size_t)c * ROWS_PER_CHUNK) * IND + i;
  float acc = 0.f;
  for (int s = 0; s < ROWS_PER_CHUNK; ++s) acc += xb[(size_t)s * IND];
  xsp[(size_t)(c * B_ + b) * IND + i] = acc;
}

// ---------------------------------------------------------------------------
// Kernel 2/4/6: per-(b,o) capsule update. One wave per (b,o).
//   yvec = (mode0 ? xsum/16 : sum_chunks ypart)      [128]
//   s_d  = W[o*32+d] . yvec ;  a = squash(s)
//   mode 0: V  = W_o^T a ;  mode 1: V += W_o^T a ;  mode 2: out = a
// ---------------------------------------------------------------------------
__global__ __launch_bounds__(32) void caps_update(const float* __restrict__ W,
                                                  const float* __restrict__ part,
                                                  float* __restrict__ Vbuf,
                                                  float* __restrict__ out,
                                                  int mode) {
  const int bo = blockIdx.x;          // b*16 + o
  const int b  = bo >> 4;
  const int o  = bo & 15;
  const int lane = threadIdx.x;       // wave32
  __shared__ float yvec[IND];

  #pragma unroll
  for (int t = 0; t < 4; ++t) {
    const int i = lane + 32 * t;
    float acc = 0.f;
    if (mode == 0) {
      for (int c = 0; c < CHUNKS; ++c) acc += part[(size_t)(c * B_ + b) * IND + i];
      acc *= (1.0f / 16.0f);          // uniform softmax at r=0
    } else {
      for (int c = 0; c < CHUNKS; ++c)
        acc += part[((size_t)(c * B_ + b) * ONUM + o) * IND + i];
    }
    yvec[i] = acc;
  }
  __syncthreads();

  // s_d = W row (o*32+lane) dot yvec
  const float* wr = W + (size_t)(o * ODIM + lane) * IND;
  float s = 0.f;
  #pragma unroll 4
  for (int i = 0; i < IND; ++i) s = fmaf(wr[i], yvec[i], s);

  // squash: a = sqrt(n)/(1+n) * s, n = |s|^2 (reduce across the 32 lanes)
  float n = s * s;
  n += __shfl_xor(n, 16, 32);
  n += __shfl_xor(n,  8, 32);
  n += __shfl_xor(n,  4, 32);
  n += __shfl_xor(n,  2, 32);
  n += __shfl_xor(n,  1, 32);
  const float a = sqrtf(n) / (1.0f + n) * s;

  if (mode == 2) {
    out[(size_t)bo * ODIM + lane] = a;
  } else {
    // V[b][o][i] (+)= sum_d a_d * W[o*32+d][i]
    float vadd[4] = {0.f, 0.f, 0.f, 0.f};
    for (int d2 = 0; d2 < ODIM; ++d2) {
      const float ad = __shfl(a, d2, 32);
      const float* w2 = W + (size_t)(o * ODIM + d2) * IND;
      #pragma unroll
      for (int t = 0; t < 4; ++t) vadd[t] = fmaf(ad, w2[lane + 32 * t], vadd[t]);
    }
    float* Vp = Vbuf + (size_t)bo * IND;
    if (mode == 0) {
      #pragma unroll
      for (int t = 0; t < 4; ++t) Vp[lane + 32 * t] = vadd[t];
    } else {
      #pragma unroll
      for (int t = 0; t < 4; ++t) Vp[lane + 32 * t] += vadd[t];
    }
  }
}

// ---------------------------------------------------------------------------
// Kernel 3/5: routing pass. Block = (batch, chunk of 128 seq rows), 4 waves.
// Per 16-row tile per wave:
//   U[16s x 16o] = Xtile[16x128] @ V^T  (32 x V_WMMA_F32_16X16X4_F32, K=128)
//   c = softmax over o (across 16-lane halves, matching C-layout)
//   y[16o x 128f] += c^T @ Xtile        (32 more WMMAs, K=16)
// Then block-reduce wave partials in LDS and store per-chunk partial (no atomics).
// ---------------------------------------------------------------------------
__global__ __launch_bounds__(128) void routing_pass(const float* __restrict__ x,
                                                    const float* __restrict__ Vbuf,
                                                    float* __restrict__ ypart) {
  __shared__ float smem[2112 + WAVES * 2112 + WAVES * (16 * CS_STRIDE)]; // 46.6 KB
  float* Vsh = smem;                                  // 16 x 132
  const int tid  = threadIdx.x;
  const int w    = tid >> 5;
  const int lane = tid & 31;
  const int half = lane >> 4;
  const int m    = lane & 15;
  float* xt  = smem + 2112 + w * 2112;                // per-wave 16 x 132 x tile
  float* csh = smem + 2112 + WAVES * 2112 + w * (16 * CS_STRIDE); // per-wave 16 x 17

  const int batch = blockIdx.x;
  const int chunk = blockIdx.y;

  // cooperative load of V[batch] (16 x 128) into LDS
  const float* Vb = Vbuf + (size_t)batch * (ONUM * IND);
  for (int idx = tid; idx < ONUM * IND; idx += 128) {
    Vsh[(idx >> 7) * XT_STRIDE + (idx & 127)] = Vb[idx];
  }
  __syncthreads();

  v8f yacc[8];
  #pragma unroll
  for (int nt = 0; nt < 8; ++nt) yacc[nt] = (v8f){0,0,0,0,0,0,0,0};

  for (int tt = 0; tt < TILES_PER_WAVE; ++tt) {
    const int t    = w + tt * WAVES;
    const int row0 = chunk * ROWS_PER_CHUNK + t * 16;

    // load this wave's 16x128 x tile into padded LDS
    const float* xb = x + ((size_t)batch * SEQ + row0) * IND;
    #pragma unroll
    for (int q = 0; q < 16; ++q) {
      const v4f val = *reinterpret_cast<const v4f*>(xb + (size_t)q * IND + lane * 4);
      *reinterpret_cast<v4f*>(&xt[q * XT_STRIDE + lane * 4]) = val;
    }
    __syncthreads();

    // ---- logits U = Xtile @ V^T : fp32 WMMA chain over K=128 ----
    // A frag (16x4, M=s): lane -> M=m, K=v+2*half ; B frag (4x16, N=o): lane -> N=m, K=v+2*half
    v8f u = (v8f){0,0,0,0,0,0,0,0};
    #pragma unroll
    for (int kk = 0; kk < 32; ++kk) {
      const int off = kk * 4 + 2 * half;
      const v2f af = *reinterpret_cast<const v2f*>(&xt [m * XT_STRIDE + off]);
      const v2f bf = *reinterpret_cast<const v2f*>(&Vsh[m * XT_STRIDE + off]);
      u = __builtin_amdgcn_wmma_f32_16x16x4_f32(false, af, false, bf, (short)0, u,
                                                false, false);
    }

    // ---- softmax over o (N dim = lanes within each 16-lane half) ----
    #pragma unroll
    for (int j = 0; j < 8; ++j) {
      float v = u[j];
      float mx = v;
      mx = fmaxf(mx, __shfl_xor(mx, 1, 32));
      mx = fmaxf(mx, __shfl_xor(mx, 2, 32));
      mx = fmaxf(mx, __shfl_xor(mx, 4, 32));
      mx = fmaxf(mx, __shfl_xor(mx, 8, 32));
      float e = __expf(v - mx);
      float sm = e;
      sm += __shfl_xor(sm, 1, 32);
      sm += __shfl_xor(sm, 2, 32);
      sm += __shfl_xor(sm, 4, 32);
      sm += __shfl_xor(sm, 8, 32);
      const float c = e / sm;
      // C layout: elem j, lanes 0-15 -> s=j ; lanes 16-31 -> s=j+8 ; o = m
      csh[(j + 8 * half) * CS_STRIDE + m] = c;
    }
    __syncthreads();

    // ---- y += c^T @ Xtile : A = c^T (M=o,K=s), B = Xtile (K=s,N=f) ----
    v2f a2[4];
    #pragma unroll
    for (int kk2 = 0; kk2 < 4; ++kk2) {
      const int k0 = kk2 * 4 + 2 * half;
      a2[kk2].x = csh[k0 * CS_STRIDE + m];
      a2[kk2].y = csh[(k0 + 1) * CS_STRIDE + m];
    }
    #pragma unroll
    for (int nt = 0; nt < 8; ++nt) {
      #pragma unroll
      for (int kk2 = 0; kk2 < 4; ++kk2) {
        const int k0 = kk2 * 4 + 2 * half;
        v2f bf;
        bf.x = xt[k0 * XT_STRIDE + nt * 16 + m];
        bf.y = xt[(k0 + 1) * XT_STRIDE + nt * 16 + m];
        yacc[nt] = __builtin_amdgcn_wmma_f32_16x16x4_f32(false, a2[kk2], false, bf,
                                                         (short)0, yacc[nt],
                                                         false, false);
      }
    }
    __syncthreads();
  }

  // ---- block reduction of per-wave y partials (reuse x-tile LDS) ----
  float* yred = smem + 2112 + w * 2112;   // 16*128 fits in 16*132 region
  #pragma unroll
  for (int nt = 0; nt < 8; ++nt) {
    #pragma unroll
    for (int r = 0; r < 8; ++r) {
      const int o = r + 8 * half;
      const int f = nt * 16 + m;
      yred[o * IND + f] = yacc[nt][r];
    }
  }
  __syncthreads();

  float* dst = ypart + ((size_t)(chunk * B_ + batch)) * (ONUM * IND);
  for (int idx = tid; idx < ONUM * IND; idx += 128) {
    const float s = smem[2112 + 0 * 2112 + idx] + smem[2112 + 1 * 2112 + idx] +
                    smem[2112 + 2 * 2112 + idx] + smem[2112 + 3 * 2112 + idx];
    dst[idx] = s;
  }
}

// ---------------------------------------------------------------------------
extern "C" void kernel_launch(void* const* d_in, const int* in_sizes, int n_in,
                              void* d_out, int out_size, void* d_ws, size_t ws_size,
                              hipStream_t stream) {
  (void)in_sizes; (void)n_in; (void)out_size; (void)ws_size;
  const float* x = (const float*)d_in[0];  // [32, 2048, 128]
  const float* W = (const float*)d_in[1];  // [512, 128, 1]
  float* out = (float*)d_out;              // [32, 16, 32]
  float* ws  = (float*)d_ws;
  float* xsum_part = ws + OFF_XSUM;        // fully rewritten each call
  float* Vbuf      = ws + OFF_V;           // mode 0 writes (no stale state)
  float* ypart     = ws + OFF_YP;          // fully rewritten each pass

  // r = 0: uniform routing -> a0, V1 = W^T a0
  xsum_kernel <<<dim3(B_, CHUNKS), 128, 0, stream>>>(x, xsum_part);
  caps_update <<<B_ * ONUM, 32, 0, stream>>>(W, xsum_part, Vbuf, out, 0);
  // r = 1
  routing_pass<<<dim3(B_, CHUNKS), 128, 0, stream>>>(x, Vbuf, ypart);
  caps_update <<<B_ * ONUM, 32, 0, stream>>>(W, ypart, Vbuf, out, 1);
  // r = 2 (final)
  routing_pass<<<dim3(B_, CHUNKS), 128, 0, stream>>>(x, Vbuf, ypart);
  caps_update <<<B_ * ONUM, 32, 0, stream>>>(W, ypart, Vbuf, out, 2);
}